// BallQuery_14456859919013
// MI455X (gfx1250) — compile-verified
//
#include <hip/hip_runtime.h>

typedef __attribute__((ext_vector_type(2))) float v2f;
typedef __attribute__((ext_vector_type(8))) float v8f;

#define N1_Q   8192
#define N2_P   32768
#define KNN    10
#define RAD2   0.0625f   // 0.25^2

// One wave handles 16 queries; each iteration scans 32 points via two
// v_wmma_f32_16x16x4_f32 (shared A operand = queries, padded K 3->4).
//   d2 = |q|^2 + |p|^2 - 2*dot   (exactly the reference's fp32 formula)
// Per accumulator VGPR a wave32 ballot gives 2 rows x 16 cols of hit bits;
// lane L<16 assembles its row's 32-bit mask (ascending point index) and pops
// bits until K hits. Wave-uniform early exit once all 16 rows are full.
__global__ __launch_bounds__(256) void ball_query_wmma_kernel(
    const float* __restrict__ pc1,   // [8192, 3]
    const float* __restrict__ pc2,   // [32768, 3]
    float* __restrict__ out)         // [N1*K] mapping ++ [N1*K*3] pts
{
    const int lane = threadIdx.x & 31;
    const int wave = threadIdx.x >> 5;
    const int n    = lane & 15;
    const bool hi  = lane >= 16;

    const int qbase = (blockIdx.x * 8 + wave) * 16;
    const int gq    = qbase + n;

    // ---- A matrix (queries), ISA 32-bit A 16x4 layout --------------------
    const float qx = pc1[gq * 3 + 0];
    const float qy = pc1[gq * 3 + 1];
    const float qz = pc1[gq * 3 + 2];
    v2f a;
    a.x = hi ? qz : qx;      // VGPR0: K=0 (lo lanes) / K=2 (hi lanes)
    a.y = hi ? 0.0f : qy;    // VGPR1: K=1 (lo lanes) / K=3 pad (hi lanes)

    // |q|^2 per query row; lane L holds it for query (L&15).
    const float qq_own = qx * qx + qy * qy + qz * qz;
    // qqv[r] = |q|^2 of row (r + 8*hi)  — matches C/D row-per-VGPR layout.
    float qqv[8];
#pragma unroll
    for (int r = 0; r < 8; ++r)
        qqv[r] = __shfl(qq_own, r + (hi ? 8 : 0), 32);

    // Per-lane row-select predicates (dataflow select, not branches).
    const int rsel = lane & 7;

    float* outMap = out;                       // N1*K floats (index values)
    float* outPts = out + (size_t)N1_Q * KNN;  // N1*K*3 floats

    int cnt = hi ? KNN : 0;            // hi lanes "done" so ballots work
    const int qrow = qbase + lane;     // query owned by lane (valid lane<16)

    for (int j0 = 0; j0 < N2_P; j0 += 32) {
        if (j0 + 32 < N2_P)  // global_prefetch_b8 for the next tile
            __builtin_prefetch(pc2 + (size_t)(j0 + 32) * 3, 0, 1);

        // ---- B matrices: points j0+n and j0+16+n (ISA 4x16 B layout) ----
        const int   pj0 = j0 + n;
        const float px0 = pc2[pj0 * 3 + 0];
        const float py0 = pc2[pj0 * 3 + 1];
        const float pz0 = pc2[pj0 * 3 + 2];
        const int   pj1 = pj0 + 16;
        const float px1 = pc2[pj1 * 3 + 0];
        const float py1 = pc2[pj1 * 3 + 1];
        const float pz1 = pc2[pj1 * 3 + 2];

        v2f b0, b1;
        b0.x = hi ? pz0 : px0;  b0.y = hi ? 0.0f : py0;
        b1.x = hi ? pz1 : px1;  b1.y = hi ? 0.0f : py1;
        const float pp0 = px0 * px0 + py0 * py0 + pz0 * pz0;
        const float pp1 = px1 * px1 + py1 * py1 + pz1 * pz1;

        // ---- cross terms via matrix core (A reused across both) ---------
        v8f c0 = {}, c1 = {};
        c0 = __builtin_amdgcn_wmma_f32_16x16x4_f32(
            false, a, false, b0, (short)0, c0, false, false);
        c1 = __builtin_amdgcn_wmma_f32_16x16x4_f32(
            false, a, false, b1, (short)0, c1, false, false);

        // ---- radius test -> ballots (2 rows x 16 cols each) -------------
        unsigned balA[8], balB[8];
#pragma unroll
        for (int r = 0; r < 8; ++r) {
            const float d2a = qqv[r] + pp0 - 2.0f * c0[r];
            const float d2b = qqv[r] + pp1 - 2.0f * c1[r];
            balA[r] = __builtin_amdgcn_ballot_w32(d2a < RAD2);
            balB[r] = __builtin_amdgcn_ballot_w32(d2b < RAD2);
        }

        // ---- lane L<16 pops hits for query row L in index order ---------
        if (!hi && cnt < KNN) {
            // Gather bal*[lane&7] via OR-accumulated cndmask selects.
            unsigned wA = 0u, wB = 0u;
#pragma unroll
            for (int r = 0; r < 8; ++r) {
                wA |= (rsel == r) ? balA[r] : 0u;
                wB |= (rsel == r) ? balB[r] : 0u;
            }
            const unsigned loHalf = (lane < 8) ? 1u : 0u;
            const unsigned ha = loHalf ? (wA & 0xffffu) : (wA >> 16);
            const unsigned hb = loHalf ? (wB & 0xffffu) : (wB >> 16);
            unsigned m32 = ha | (hb << 16);   // bit b -> point index j0+b
            while (m32 && cnt < KNN) {
                const int bit = __builtin_ctz(m32);
                m32 &= (m32 - 1u);
                const int idx  = j0 + bit;
                const int slot = qrow * KNN + cnt;
                outMap[slot] = (float)idx;                 // exact (< 2^24)
                outPts[(size_t)slot * 3 + 0] = pc2[idx * 3 + 0];
                outPts[(size_t)slot * 3 + 1] = pc2[idx * 3 + 1];
                outPts[(size_t)slot * 3 + 2] = pc2[idx * 3 + 2];
                ++cnt;
            }
        }

        // uniform early exit: all 16 rows full (hi lanes start at K)
        if (__builtin_amdgcn_ballot_w32(cnt >= KNN) == 0xffffffffu)
            break;
    }

    // fill unmatched slots: mapping = -1, pts = 0
    if (!hi) {
        for (int s = cnt; s < KNN; ++s) {
            const int slot = qrow * KNN + s;
            outMap[slot] = -1.0f;
            outPts[(size_t)slot * 3 + 0] = 0.0f;
            outPts[(size_t)slot * 3 + 1] = 0.0f;
            outPts[(size_t)slot * 3 + 2] = 0.0f;
        }
    }
}

extern "C" void kernel_launch(void* const* d_in, const int* in_sizes, int n_in,
                              void* d_out, int out_size, void* d_ws, size_t ws_size,
                              hipStream_t stream) {
    (void)in_sizes; (void)n_in; (void)d_ws; (void)ws_size; (void)out_size;
    const float* pc1 = (const float*)d_in[0];   // (1, 8192, 3)  fp32
    const float* pc2 = (const float*)d_in[1];   // (1, 32768, 3) fp32
    float* out = (float*)d_out;                 // 8192*10 + 8192*10*3 floats

    // 64 blocks x 8 waves x 16 queries = 8192 queries
    ball_query_wmma_kernel<<<dim3(N1_Q / 128), dim3(256), 0, stream>>>(pc1, pc2, out);
}